// EdgeConv_11476152615283
// MI455X (gfx1250) — compile-verified
//
#include <hip/hip_runtime.h>
#include <cstdint>

typedef float v2f __attribute__((ext_vector_type(2)));
typedef float v8f __attribute__((ext_vector_type(8)));

#define B_ 16
#define C_ 64
#define N_ 2048
#define K_ 20
#define OUT_ 64
#define NEG_SLOPE 0.2f
#define BN_EPS 1e-5f

// workspace layout (byte offsets)
#define OFF_XX    0u         // B*N floats            (131072 B)
#define OFF_STATS 131072u    // 128 floats            (512 B)
#define OFF_IDX   131584u    // B*N*K ints            (2621440 B)
#define OFF_Y1    2753024u   // B*N*64 floats, [b][n][o] (8 MB)
#define OFF_Y2    11141632u  // B*N*64 floats, [b][n][o] (8 MB)
#define OFF_M     19530240u  // B*N*64 floats, [b][n][o] (8 MB)

// ---------------- Kernel A: xx[b][n] = sum_c x^2, and zero stats ----------------
__global__ __launch_bounds__(256) void k_xx(const float* __restrict__ x,
                                            float* __restrict__ xx,
                                            float* __restrict__ stats) {
  if (blockIdx.x == 0 && threadIdx.x < 128) stats[threadIdx.x] = 0.f;
  int t = blockIdx.x * 256 + threadIdx.x;           // t = b*N + n
  int b = t >> 11, n = t & 2047;
  const float* xp = x + (size_t)b * (C_ * N_) + n;
  float a = 0.f;
#pragma unroll
  for (int c = 0; c < C_; ++c) { float v = xp[c * N_]; a += v * v; }
  xx[t] = a;
}

// ---------------- Kernel B: WMMA Gram tile + top-K selection ----------------
// grid (N/16, B), 256 threads (8 waves). 16 rows x 2048 cols neg-distance in LDS.
__global__ __launch_bounds__(256) void k_knn(const float* __restrict__ x,
                                             const float* __restrict__ xx,
                                             int* __restrict__ idxg) {
  __shared__ float As[C_ * 16];     // A tile [k][m], 4 KB
  __shared__ float xxr[16];
  __shared__ float dist[16 * N_];   // 128 KB neg-distance tile
  const int n0 = blockIdx.x * 16;
  const int b  = blockIdx.y;
  const int t  = threadIdx.x;
  const int lane = t & 31, w = t >> 5;
  const float* xb = x + (size_t)b * (C_ * N_);

  // stage A tile: As[k*16+m] = x[b,k,n0+m]
  for (int i = 0; i < 4; ++i) {
    int l = t * 4 + i; int k = l >> 4, m = l & 15;
    As[l] = xb[k * N_ + n0 + m];
  }
  if (t < 16) xxr[t] = xx[b * N_ + n0 + t];
  __syncthreads();

  const int mrow  = lane & 15;
  const int khalf = (lane >> 4) << 1;   // lanes 16-31 carry K+2,K+3 (ISA A/B layout)
  const int mbase = (lane >> 4) << 3;   // C/D: upper lanes hold M=8..15

  // each wave covers 256 columns = 16 subtiles of 16
  for (int ct = 0; ct < 16; ++ct) {
    int col = w * 256 + ct * 16 + mrow;
    v8f acc = {};
#pragma unroll
    for (int kk = 0; kk < 16; ++kk) {     // K=64 in chunks of 4
      int kb = kk * 4 + khalf;
      v2f a, bb;
      a.x  = As[(kb + 0) * 16 + mrow];
      a.y  = As[(kb + 1) * 16 + mrow];
      bb.x = xb[(kb + 0) * N_ + col];
      bb.y = xb[(kb + 1) * N_ + col];
      acc = __builtin_amdgcn_wmma_f32_16x16x4_f32(false, a, false, bb,
                                                  (short)0, acc, false, false);
    }
    float xxc = xx[b * N_ + col];
#pragma unroll
    for (int r = 0; r < 8; ++r) {
      int m = mbase + r;                  // neg_d = 2*G - |xi|^2 - |xj|^2
      dist[m * N_ + col] = 2.0f * acc[r] - xxr[m] - xxc;
    }
  }
  __syncthreads();

  // top-K: wave w handles rows 2w, 2w+1; 20x (wave argmax + invalidate)
  for (int rr = 0; rr < 2; ++rr) {
    int m = w * 2 + rr;
    float* drow = &dist[m * N_];
    int* orow = idxg + ((size_t)(b * N_ + n0 + m)) * K_;
    for (int sel = 0; sel < K_; ++sel) {
      float bv = -3.0e38f; int bj = 0x7fffffff;
      for (int tt = 0; tt < 64; ++tt) {
        int j = tt * 32 + lane;           // conflict-free LDS banks
        float v = drow[j];
        if (v > bv) { bv = v; bj = j; }
      }
#pragma unroll
      for (int off = 16; off > 0; off >>= 1) {
        float ov = __shfl_xor(bv, off, 32);
        int   oj = __shfl_xor(bj, off, 32);
        if (ov > bv || (ov == bv && oj < bj)) { bv = ov; bj = oj; }
      }
      if (lane == 0) { orow[sel] = bj; drow[bj] = -3.0e38f; }
    }
  }
}

// ---------------- Kernel C: Y1 = (W1-W2)*x, Y2 = W2*x, stored [b][n][o] ----------------
// grid (16 n-chunks of 128, B), 256 threads (8 waves x 16 cols).
__global__ __launch_bounds__(256) void k_transform(const float* __restrict__ x,
                                                   const float* __restrict__ W,
                                                   float* __restrict__ Y1,
                                                   float* __restrict__ Y2) {
  __shared__ float Aks[C_ * OUT_];  // [k][o], 16 KB
  __shared__ float S[128 * OUT_];   // [n_local][o], 32 KB
  const int nchunk = blockIdx.x, b = blockIdx.y;
  const int t = threadIdx.x, lane = t & 31, w = t >> 5;
  const float* xb = x + (size_t)b * (C_ * N_);
  const int mrow  = lane & 15;
  const int khalf = (lane >> 4) << 1;
  const int obump = (lane >> 4) << 3;

  for (int pass = 0; pass < 2; ++pass) {
    for (int l = t; l < C_ * OUT_; l += 256) {
      int k = l >> 6, o = l & 63;
      float w2 = W[o * (2 * C_) + C_ + k];
      Aks[l] = pass ? w2 : (W[o * (2 * C_) + k] - w2);
    }
    __syncthreads();
    int col = nchunk * 128 + w * 16 + mrow;
    for (int ot = 0; ot < 4; ++ot) {
      v8f acc = {};
#pragma unroll
      for (int kk = 0; kk < 16; ++kk) {
        int kb = kk * 4 + khalf;
        v2f a, bb;
        a.x  = Aks[(kb + 0) * 64 + ot * 16 + mrow];
        a.y  = Aks[(kb + 1) * 64 + ot * 16 + mrow];
        bb.x = xb[(kb + 0) * N_ + col];
        bb.y = xb[(kb + 1) * N_ + col];
        acc = __builtin_amdgcn_wmma_f32_16x16x4_f32(false, a, false, bb,
                                                    (short)0, acc, false, false);
      }
      int nl = w * 16 + mrow;
#pragma unroll
      for (int r = 0; r < 8; ++r) S[nl * 64 + ot * 16 + obump + r] = acc[r];
    }
    __syncthreads();
    float* dst = (pass ? Y2 : Y1) + (size_t)b * (N_ * OUT_) + (size_t)nchunk * 128 * OUT_;
    for (int l = t; l < 128 * OUT_; l += 256) dst[l] = S[l];  // fully coalesced
    __syncthreads();
  }
}

// ---------------- Kernel D: gather, BN stats, max over neighbors ----------------
// grid 1024, 256 threads; each wave owns 4 points; lane owns channel pair (2*lane, 2*lane+1).
__global__ __launch_bounds__(256) void k_stats(const float* __restrict__ Y1,
                                               const float* __restrict__ Y2,
                                               const int* __restrict__ idxg,
                                               float* __restrict__ Mo,
                                               float* __restrict__ stats) {
  __shared__ float sred[8][64];
  __shared__ float qred[8][64];
  const int t = threadIdx.x, lane = t & 31, w = t >> 5;
  float sx = 0.f, sy = 0.f, qx = 0.f, qy = 0.f;
  int p0 = (blockIdx.x * 8 + w) * 4;
  for (int pp = 0; pp < 4; ++pp) {
    int p = p0 + pp;
    int b = p >> 11;
    const float2 y1 = *(const float2*)(Y1 + (size_t)p * 64 + lane * 2);
    float mx = -3.0e38f, my = -3.0e38f;
    const int* ip = idxg + (size_t)p * K_;
    for (int k = 0; k < K_; ++k) {
      int m = ip[k];
      const float2 q = *(const float2*)(Y2 + ((size_t)(b << 11) + m) * 64 + lane * 2);
      float hx = y1.x + q.x, hy = y1.y + q.y;
      sx += hx; sy += hy; qx += hx * hx; qy += hy * hy;
      mx = fmaxf(mx, q.x); my = fmaxf(my, q.y);
    }
    float2 mo; mo.x = y1.x + mx; mo.y = y1.y + my;   // max commutes (scale>0)
    *(float2*)(Mo + (size_t)p * 64 + lane * 2) = mo;
  }
  sred[w][lane * 2] = sx; sred[w][lane * 2 + 1] = sy;
  qred[w][lane * 2] = qx; qred[w][lane * 2 + 1] = qy;
  __syncthreads();
  if (t < 64) {
    float s = 0.f;
#pragma unroll
    for (int i = 0; i < 8; ++i) s += sred[i][t];
    __hip_atomic_fetch_add(&stats[t], s, __ATOMIC_RELAXED, __HIP_MEMORY_SCOPE_AGENT);
  } else if (t < 128) {
    int o = t - 64;
    float s = 0.f;
#pragma unroll
    for (int i = 0; i < 8; ++i) s += qred[i][o];
    __hip_atomic_fetch_add(&stats[64 + o], s, __ATOMIC_RELAXED, __HIP_MEMORY_SCOPE_AGENT);
  }
}

// ---------------- Kernel E: BN affine + LeakyReLU + transpose to (B,OUT,N) ----------------
__global__ __launch_bounds__(256) void k_final(const float* __restrict__ Mo,
                                               const float* __restrict__ stats,
                                               const float* __restrict__ gamma,
                                               const float* __restrict__ beta,
                                               float* __restrict__ out) {
  __shared__ float T[64 * 64];
  __shared__ float sc[64], bs[64];
  const int nt = blockIdx.x, b = blockIdx.y, t = threadIdx.x;
  if (t < 64) {
    const float invCnt = 1.0f / (float)(B_ * N_ * K_);
    float mean = stats[t] * invCnt;
    float var  = stats[64 + t] * invCnt - mean * mean;
    float s    = rsqrtf(var + BN_EPS) * gamma[t];
    sc[t] = s;
    bs[t] = beta[t] - mean * s;
  }
  __syncthreads();
  const int n0 = nt * 64;
  for (int l = t; l < 4096; l += 256) {
    int nl = l >> 6, o = l & 63;
    float v = Mo[((size_t)(b * N_ + n0 + nl)) * 64 + o] * sc[o] + bs[o];
    T[l] = v >= 0.f ? v : NEG_SLOPE * v;
  }
  __syncthreads();
  for (int l = t; l < 4096; l += 256) {
    int o = l >> 6, nl = l & 63;
    out[(size_t)b * (OUT_ * N_) + (size_t)o * N_ + n0 + nl] = T[nl * 64 + o];
  }
}

extern "C" void kernel_launch(void* const* d_in, const int* in_sizes, int n_in,
                              void* d_out, int out_size, void* d_ws, size_t ws_size,
                              hipStream_t stream) {
  (void)in_sizes; (void)n_in; (void)out_size; (void)ws_size;
  const float* x     = (const float*)d_in[0];
  const float* W     = (const float*)d_in[1];
  const float* gamma = (const float*)d_in[2];
  const float* beta  = (const float*)d_in[3];
  char* ws = (char*)d_ws;
  float* xx    = (float*)(ws + OFF_XX);
  float* stats = (float*)(ws + OFF_STATS);
  int*   idxg  = (int*)(ws + OFF_IDX);
  float* Y1    = (float*)(ws + OFF_Y1);
  float* Y2    = (float*)(ws + OFF_Y2);
  float* Mo    = (float*)(ws + OFF_M);
  float* out   = (float*)d_out;

  k_xx       <<<128,            256, 0, stream>>>(x, xx, stats);
  k_knn      <<<dim3(128, 16),  256, 0, stream>>>(x, xx, idxg);
  k_transform<<<dim3(16, 16),   256, 0, stream>>>(x, W, Y1, Y2);
  k_stats    <<<1024,           256, 0, stream>>>(Y1, Y2, idxg, Mo, stats);
  k_final    <<<dim3(32, 16),   256, 0, stream>>>(Mo, stats, gamma, beta, out);
}